// VectorQuantizer_70102456205879
// MI455X (gfx1250) — compile-verified
//
#include <hip/hip_runtime.h>

// ---------------------------------------------------------------------------
// VQ quantizer for MI455X (gfx1250, wave32, WMMA).
//
// Shapes: z [8,4096,256] f32, codebook [4096,256] f32.
// d2 GEMM = 68.7 GFLOP; minimal HBM traffic ~70MB (3us @ 23.3TB/s) => compute
// bound IF the 512MB d2 matrix never hits HBM. We exploit CDNA5's 320KB LDS:
// a 16-row x 4096-code fp32 d2 stripe (256KB) stays LDS-resident per WGP, so
// the GEMM runs exactly once and all reductions (argmin / softmax / avg_probs
// / gather / MSE) read LDS. fp32 WMMA (v_wmma_f32_16x16x4_f32) preserves the
// reference's argmin fidelity (code-to-code d2 gaps ~1e-5 would be destroyed
// by bf16 operands).
// ---------------------------------------------------------------------------

#define N_TOK   32768      // B*L
#define DDIM    256
#define KCODES  4096
#define ROWS    16         // token rows per workgroup
#define ZSTR    260        // padded LDS row stride (floats) -> conflict-free A reads
#define TPB     256
#define WAVES   8

typedef float v2f __attribute__((ext_vector_type(2)));
typedef float v8f __attribute__((ext_vector_type(8)));

// LDS layout (floats)
#define OFF_ZA     0
#define OFF_D2     (ROWS * ZSTR)                  // 4160
#define OFF_RMIN   (OFF_D2 + ROWS * KCODES)       // 69696
#define OFF_RINVZ  (OFF_RMIN + ROWS)
#define OFF_RIDX   (OFF_RINVZ + ROWS)
#define OFF_ZNORM  (OFF_RIDX + ROWS)
#define OFF_WPART  (OFF_ZNORM + ROWS)
#define SMEM_FLOATS (OFF_WPART + WAVES)           // 69768 floats = 279072 B <= 320KB

// ---------------------------------------------------------------------------
// Kernel 1: per-code squared norms + zero global accumulators.
// ---------------------------------------------------------------------------
__global__ __launch_bounds__(TPB) void vq_init(const float* __restrict__ cb,
                                               float* __restrict__ cnorm,
                                               float* __restrict__ avg_acc,
                                               float* __restrict__ mse_acc) {
  int k = blockIdx.x * TPB + threadIdx.x;
  if (k < KCODES) {
    const float4* row = (const float4*)(cb + (size_t)k * DDIM);
    float s = 0.f;
#pragma unroll 8
    for (int i = 0; i < DDIM / 4; ++i) {
      float4 v = row[i];
      s += v.x * v.x + v.y * v.y + v.z * v.z + v.w * v.w;
    }
    cnorm[k] = s;
    avg_acc[k] = 0.f;
    if (k == 0) *mse_acc = 0.f;
  }
}

// ---------------------------------------------------------------------------
// Kernel 2: fused distance GEMM (WMMA f32) + LDS-resident reductions.
// One workgroup = 16 token rows x all 4096 codes.
// ---------------------------------------------------------------------------
__global__ __launch_bounds__(TPB, 1) void vq_main(const float* __restrict__ z,
                                                  const float* __restrict__ cb,
                                                  const float* __restrict__ cnorm,
                                                  float* __restrict__ zq_out,
                                                  int* __restrict__ idx_out,
                                                  float* __restrict__ avg_acc,
                                                  float* __restrict__ mse_acc) {
  extern __shared__ float smem[];
  float* zA   = smem + OFF_ZA;
  float* d2s  = smem + OFF_D2;
  float* rmin = smem + OFF_RMIN;
  float* rinv = smem + OFF_RINVZ;
  int*   ridx = (int*)(smem + OFF_RIDX);
  float* znrm = smem + OFF_ZNORM;
  float* wprt = smem + OFF_WPART;

  const int tid  = threadIdx.x;
  const int lane = tid & 31;
  const int wave = tid >> 5;
  const size_t row0 = (size_t)blockIdx.x * ROWS;

  // ---- stage z tile [16 x 256] into padded LDS ----
  {
    const float4* zg = (const float4*)(z + row0 * DDIM);
#pragma unroll
    for (int i = 0; i < 4; ++i) {
      int lin = tid + i * TPB;           // 1024 float4s
      int r = lin >> 6;                  // 64 float4 per row
      int c = lin & 63;
      float4 v = zg[(size_t)r * 64 + c];
      *(float4*)(zA + r * ZSTR + c * 4) = v;  // row base r*1040B is 16B aligned
    }
  }
  __syncthreads();
  if (tid < ROWS) {
    const float* zr = zA + tid * ZSTR;
    float s = 0.f;
#pragma unroll 8
    for (int d = 0; d < DDIM; ++d) s += zr[d] * zr[d];
    znrm[tid] = s;
  }
  __syncthreads();

  // ---- GEMM: d2[r][k] = |z_r|^2 + |c_k|^2 - 2 * z_r . c_k ----
  // A frag (16x4 f32): lane l -> row l&15; VGPR0/1 hold K = 2*(l>>4) + {0,1}.
  // B frag (4x16 f32): lane l -> col l&15; VGPR0/1 hold K = 2*(l>>4) + {0,1}.
  const int rA = lane & 15;
  const int kb = (lane >> 4) * 2;
  const float* aBase = zA + rA * ZSTR + kb;
  const int cC = lane & 15;
  const int rC = (lane >> 4) * 8;

#pragma unroll 1
  for (int t2 = wave; t2 < 128; t2 += WAVES) {  // each wave: 16 tile-pairs
    const int k0 = t2 * 32;                     // two adjacent 16-code tiles
    const float* b0 = cb + ((size_t)(k0)      + rA) * DDIM + kb;
    const float* b1 = cb + ((size_t)(k0 + 16) + rA) * DDIM + kb;
    // hint next pair of codebook tiles toward the caches
    if (t2 + WAVES < 128)
      __builtin_prefetch(cb + ((size_t)(k0 + WAVES * 32) + rA) * DDIM + kb, 0, 1);

    v8f acc0 = {};
    v8f acc1 = {};
#pragma unroll 8
    for (int d0 = 0; d0 < DDIM; d0 += 4) {
      v2f a;  a.x  = aBase[d0]; a.y  = aBase[d0 + 1];   // ds_load_b64
      v2f bA; bA.x = b0[d0];    bA.y = b0[d0 + 1];      // global_load_b64 (L2 hit)
      v2f bB; bB.x = b1[d0];    bB.y = b1[d0 + 1];
      acc0 = __builtin_amdgcn_wmma_f32_16x16x4_f32(false, a, false, bA,
                                                   (short)0, acc0, false, false);
      acc1 = __builtin_amdgcn_wmma_f32_16x16x4_f32(false, a, false, bB,
                                                   (short)0, acc1, false, false);
    }
    // epilogue: C/D layout -> lane l owns col l&15, rows 8*(l>>4)+v
#pragma unroll
    for (int v = 0; v < 8; ++v) {
      const int rr = rC + v;
      const float zn = znrm[rr];
      d2s[rr * KCODES + k0 + cC]      = zn + cnorm[k0 + cC]      - 2.f * acc0[v];
      d2s[rr * KCODES + k0 + 16 + cC] = zn + cnorm[k0 + 16 + cC] - 2.f * acc1[v];
    }
  }
  __syncthreads();

  // ---- per-row argmin (first-index tie-break) + shifted logsumexp ----
  for (int rr = wave; rr < ROWS; rr += WAVES) {
    const float* row = d2s + rr * KCODES;
    float m = 3.0e38f; int mi = 0;
    for (int k = lane; k < KCODES; k += 32) {
      float v = row[k];
      if (v < m) { m = v; mi = k; }
    }
#pragma unroll
    for (int off = 16; off > 0; off >>= 1) {
      float om = __shfl_xor(m, off, 32);
      int   oi = __shfl_xor(mi, off, 32);
      if (om < m || (om == m && oi < mi)) { m = om; mi = oi; }
    }
    float s = 0.f;
    for (int k = lane; k < KCODES; k += 32) s += __expf(m - row[k]);
#pragma unroll
    for (int off = 16; off > 0; off >>= 1) s += __shfl_xor(s, off, 32);
    if (lane == 0) { rmin[rr] = m; ridx[rr] = mi; rinv[rr] = 1.f / s; }
  }
  __syncthreads();

  // ---- avg_probs[k] += sum_r softmax(-d2)[r][k]  (consecutive k -> no bank conflicts)
  for (int k = tid; k < KCODES; k += TPB) {
    float s = 0.f;
#pragma unroll
    for (int rr = 0; rr < ROWS; ++rr)
      s += __expf(rmin[rr] - d2s[rr * KCODES + k]) * rinv[rr];
    atomicAdd(&avg_acc[k], s);   // global_atomic_add_f32, no-return
  }

  // ---- gather z_q (== z_q_ste forward value), indices, MSE partial ----
  {
    const int r   = tid >> 4;
    const int sub = tid & 15;
    const int idx = ridx[r];
    const float4* q4 = (const float4*)(cb + (size_t)idx * DDIM);
    float4*       o4 = (float4*)(zq_out + (row0 + r) * DDIM);
    const float4* z4 = (const float4*)(zA + r * ZSTR);
    float msep = 0.f;
#pragma unroll
    for (int i = 0; i < 4; ++i) {
      int c = sub + i * 16;
      float4 q = q4[c];
      o4[c] = q;
      float4 zv = z4[c];
      float dx = q.x - zv.x, dy = q.y - zv.y, dz = q.z - zv.z, dw = q.w - zv.w;
      msep += dx * dx + dy * dy + dz * dz + dw * dw;
    }
    if (tid < ROWS) idx_out[row0 + tid] = ridx[tid];
#pragma unroll
    for (int off = 16; off > 0; off >>= 1) msep += __shfl_xor(msep, off, 32);
    if (lane == 0) wprt[wave] = msep;
  }
  __syncthreads();
  if (tid == 0) {
    float t = 0.f;
#pragma unroll
    for (int w = 0; w < WAVES; ++w) t += wprt[w];
    atomicAdd(mse_acc, t);
  }
}

// ---------------------------------------------------------------------------
// Kernel 3: entropy over avg_probs + scalar losses.
// Forward values: codebook_loss == commitment_loss == MSE(z_q, z).
// ---------------------------------------------------------------------------
__global__ __launch_bounds__(TPB) void vq_final(const float* __restrict__ avg_acc,
                                                const float* __restrict__ mse_acc,
                                                float* __restrict__ scal) {
  __shared__ float wp[WAVES];
  const int tid = threadIdx.x, lane = tid & 31, wave = tid >> 5;
  float ent = 0.f;
  for (int k = tid; k < KCODES; k += TPB) {
    float p = avg_acc[k] * (1.0f / (float)N_TOK);
    ent -= p * __logf(p + 1e-10f);
  }
#pragma unroll
  for (int off = 16; off > 0; off >>= 1) ent += __shfl_xor(ent, off, 32);
  if (lane == 0) wp[wave] = ent;
  __syncthreads();
  if (tid == 0) {
    float e = 0.f;
#pragma unroll
    for (int w = 0; w < WAVES; ++w) e += wp[w];
    float mse = *mse_acc * (1.0f / ((float)N_TOK * (float)DDIM));
    float ccl = mse * (1.0f + 0.25f);   // codebook + COMMITMENT_COST*commitment
    float sel = 0.1f * (-e);            // ENTROPY_WEIGHT * (-entropy)
    scal[0] = ccl + sel;                // vq_loss
    scal[1] = ccl;                      // codebook_commitment_loss
    scal[2] = sel;                      // scaled_entropy_loss
  }
}

// ---------------------------------------------------------------------------
// d_out layout (reference return order, flattened):
//   [0, 8388608)            z_q_ste  f32
//   [8388608, 8421376)      indices  (written as raw int32)
//   [8421376, 8421379)      vq_loss, codebook_commitment_loss, scaled_entropy_loss
// d_ws layout (floats): [0,4096) cnorm | [4096,8192) avg_probs sum | [8192] mse sum
// ---------------------------------------------------------------------------
extern "C" void kernel_launch(void* const* d_in, const int* in_sizes, int n_in,
                              void* d_out, int out_size, void* d_ws, size_t ws_size,
                              hipStream_t stream) {
  (void)in_sizes; (void)n_in; (void)out_size; (void)ws_size;
  const float* z  = (const float*)d_in[0];
  const float* cb = (const float*)d_in[1];

  float* ws      = (float*)d_ws;
  float* cnorm   = ws;                 // 4096
  float* avg_acc = ws + KCODES;        // 4096
  float* mse_acc = ws + 2 * KCODES;    // 1

  float* zq_out  = (float*)d_out;
  int*   idx_out = (int*)((float*)d_out + (size_t)N_TOK * DDIM);
  float* scal    = (float*)d_out + (size_t)N_TOK * DDIM + N_TOK;

  vq_init<<<KCODES / TPB, TPB, 0, stream>>>(cb, cnorm, avg_acc, mse_acc);

  const size_t smem = (size_t)SMEM_FLOATS * sizeof(float);  // ~273KB (<=320KB/WGP)
  vq_main<<<N_TOK / ROWS, TPB, smem, stream>>>(z, cb, cnorm, zq_out, idx_out,
                                               avg_acc, mse_acc);

  vq_final<<<1, TPB, 0, stream>>>(avg_acc, mse_acc, scal);
}